// SemiConvModel_19447611916702
// MI455X (gfx1250) — compile-verified
//
#include <hip/hip_runtime.h>

#define CIN   8
#define COUT  16
#define HWD   192
#define TILE  32
#define HALO  2
#define TIN   36            // TILE + 2*HALO
#define PLANE (TIN * TIN)   // 1296 floats per channel plane

// Max-plus separable 5x5 conv, CDNA5 (gfx1250), wave32.
// Grid: (36 tiles, 4 batch, 4 co-groups). Block: 256 threads (8 waves).
__global__ __launch_bounds__(256)
void semiconv_maxplus_kernel(const float* __restrict__ x,
                             const float* __restrict__ scales,
                             float* __restrict__ out) {
    __shared__ float sTile[CIN * PLANE];   // 41472 B haloed input tile
    __shared__ float sW[4 * CIN * 2];      // {w1, w4} per (co-in-group, ci)

    const int tid    = threadIdx.x;
    const int tileId = blockIdx.x;         // 0..35  (6x6 tiles of 32x32)
    const int b      = blockIdx.y;         // 0..3
    const int cog    = blockIdx.z;         // 0..3  (group of 4 output channels)
    const int ty0    = (tileId / 6) * TILE;
    const int tx0    = (tileId % 6) * TILE;

    // ---- 1) weights: w(j) = -zj^2/(4s), row = (w4, w1, 0, w1, w4) ----------
    if (tid < 4 * CIN) {
        const int cu = tid >> 3;           // co within group
        const int ci = tid & 7;
        const int co = cog * 4 + cu;
        const float s = scales[co * CIN + ci];
        sW[tid * 2 + 0] = -1.0f / (4.0f * s);   // w1  (z^2 = 1)
        sW[tid * 2 + 1] = -4.0f / (4.0f * s);   // w4  (z^2 = 4)
    }

    // ---- 2) stage haloed tile into LDS via async global->LDS copies -------
    const float* xb = x + (size_t)b * CIN * HWD * HWD;
    for (int idx = tid; idx < CIN * PLANE; idx += 256) {
        const int ci  = idx / PLANE;
        const int rem = idx - ci * PLANE;
        const int r   = rem / TIN;
        const int c   = rem - r * TIN;
        const int gy  = ty0 + r - HALO;
        const int gx  = tx0 + c - HALO;
        if ((unsigned)gy < (unsigned)HWD && (unsigned)gx < (unsigned)HWD) {
            const float* g = xb + ((size_t)ci * HWD + gy) * HWD + gx;
            // low 32 bits of a generic LDS pointer = LDS byte offset
            unsigned laddr = (unsigned)(unsigned long long)&sTile[idx];
            asm volatile("global_load_async_to_lds_b32 %0, %1, off"
                         :: "v"(laddr), "v"(g) : "memory");
        } else {
            sTile[idx] = -__builtin_inff();    // -inf padding
        }
    }
    asm volatile("s_wait_asynccnt 0" ::: "memory");
    __syncthreads();

    // ---- 3) separable max-plus: horizontal then vertical -------------------
    const int px = tid & 31;          // tile-local x
    const int ry = (tid >> 5) * 4;    // tile-local base row of 4 owned pixels

    float acc[4][4];                  // [cu][k]
#pragma unroll
    for (int cu = 0; cu < 4; ++cu)
#pragma unroll
        for (int k = 0; k < 4; ++k) acc[cu][k] = 0.0f;

    for (int ci = 0; ci < CIN; ++ci) {
        // Weight-free pre-reductions per row, shared across all 4 co's:
        // A = center tap, B = max(+-1), C = max(+-2)
        float A[8], B[8], C[8];
        const int base = ci * PLANE + ry * TIN + px;
#pragma unroll
        for (int rr = 0; rr < 8; ++rr) {
            const float t0 = sTile[base + rr * TIN + 0];
            const float t1 = sTile[base + rr * TIN + 1];
            const float t2 = sTile[base + rr * TIN + 2];
            const float t3 = sTile[base + rr * TIN + 3];
            const float t4 = sTile[base + rr * TIN + 4];
            A[rr] = t2;
            B[rr] = fmaxf(t1, t3);
            C[rr] = fmaxf(t0, t4);
        }
#pragma unroll
        for (int cu = 0; cu < 4; ++cu) {
            const float w1 = sW[(cu * CIN + ci) * 2 + 0];
            const float w4 = sW[(cu * CIN + ci) * 2 + 1];
            float h[8];
#pragma unroll
            for (int rr = 0; rr < 8; ++rr)           // horizontal 1-D pass
                h[rr] = fmaxf(A[rr], fmaxf(B[rr] + w1, C[rr] + w4));
#pragma unroll
            for (int k = 0; k < 4; ++k) {            // vertical 1-D pass
                const float v =
                    fmaxf(h[k + 2],
                          fmaxf(fmaxf(h[k + 1], h[k + 3]) + w1,
                                fmaxf(h[k + 0], h[k + 4]) + w4));
                acc[cu][k] += v;                     // sum over ci (ascending)
            }
        }
    }

    // ---- 4) write-once output: nontemporal stores --------------------------
    float* outb = out + (size_t)b * COUT * HWD * HWD;
    const int ox = tx0 + px;
#pragma unroll
    for (int cu = 0; cu < 4; ++cu) {
        const int co = cog * 4 + cu;
#pragma unroll
        for (int k = 0; k < 4; ++k) {
            const int oy = ty0 + ry + k;
            __builtin_nontemporal_store(acc[cu][k],
                                        &outb[((size_t)co * HWD + oy) * HWD + ox]);
        }
    }
}

extern "C" void kernel_launch(void* const* d_in, const int* in_sizes, int n_in,
                              void* d_out, int out_size, void* d_ws, size_t ws_size,
                              hipStream_t stream) {
    (void)in_sizes; (void)n_in; (void)d_ws; (void)ws_size; (void)out_size;
    const float* x      = (const float*)d_in[0];   // (4, 8, 192, 192) f32
    const float* scales = (const float*)d_in[1];   // (16, 8) f32
    float*       out    = (float*)d_out;           // (4, 16, 192, 192) f32

    dim3 grid(36, 4, 4);   // 6x6 tiles, 4 batches, 4 output-channel groups
    dim3 block(256);       // 8 wave32 per workgroup
    semiconv_maxplus_kernel<<<grid, block, 0, stream>>>(x, scales, out);
}